// TopoLoss_283467841730
// MI455X (gfx1250) — compile-verified
//
#include <hip/hip_runtime.h>
#include <hip/hip_bf16.h>

#define NUM_CLASSES 10
#define B_ROWS 4096
#define D_DIM 768
#define LAM_TALS 0.5f
#define LAM_TAML 0.5f

typedef float v2f __attribute__((ext_vector_type(2)));
typedef float v8f __attribute__((ext_vector_type(8)));

// ---- workspace layout (in floats) ----
#define WS_CE    0       // scalar accumulator: sum of -logp
#define WS_TAML  1       // scalar accumulator: weighted relu-margin sum
#define WS_TALS  2       // tals result
#define WS_CNT   16      // [10]  per-class counts (float)
#define WS_CSUM  32      // [10*768] centroid sums
#define WS_EDIST 7712    // [100] pairwise centroid distances
#define WS_INVN  8192    // [4096] per-row 1/max(||emb||,1e-12)
#define WS_ZERO_N 7712   // zero [0, 7712)

static_assert(D_DIM % 16 == 0, "k-loop step");

// ---------------- zero accumulators ----------------
__global__ void k_zero(float* __restrict__ ws, int n) {
    int i = blockIdx.x * blockDim.x + threadIdx.x;
    if (i < n) ws[i] = 0.0f;
}

// ---------------- cross entropy + class counts ----------------
__global__ void k_ce_counts(const float* __restrict__ logits,
                            const int* __restrict__ labels,
                            float* __restrict__ ws) {
    int r = blockIdx.x * blockDim.x + threadIdx.x;
    if (r >= B_ROWS) return;
    const float* x = logits + (size_t)r * NUM_CLASSES;
    float m = x[0];
#pragma unroll
    for (int c = 1; c < NUM_CLASSES; ++c) m = fmaxf(m, x[c]);
    float s = 0.0f;
#pragma unroll
    for (int c = 0; c < NUM_CLASSES; ++c) s += __expf(x[c] - m);
    int lab = labels[r];
    float logp = x[lab] - m - __logf(s);
    atomicAdd(&ws[WS_CE], -logp);
    atomicAdd(&ws[WS_CNT + lab], 1.0f);
}

// ---------------- per-row inverse L2 norms (one wave32 per row) ----------------
__global__ void k_invnorm(const float* __restrict__ emb, float* __restrict__ ws) {
    int wid  = (blockIdx.x * blockDim.x + threadIdx.x) >> 5;
    int lane = threadIdx.x & 31;
    if (wid >= B_ROWS) return;
    const float* e = emb + (size_t)wid * D_DIM;
    float s = 0.0f;
    for (int c = lane; c < D_DIM; c += 32) { float v = e[c]; s += v * v; }
#pragma unroll
    for (int off = 16; off; off >>= 1) s += __shfl_xor(s, off, 32);
    if (lane == 0) ws[WS_INVN + wid] = 1.0f / fmaxf(sqrtf(s), 1e-12f);
}

// ---------------- centroid segment-sums, LDS-privatized ----------------
__global__ void k_centsum(const float* __restrict__ emb,
                          const int* __restrict__ labels,
                          float* __restrict__ ws) {
    __shared__ float lds[NUM_CLASSES * 256];
    int tid = threadIdx.x;
    int colChunk = blockIdx.x % 3;
    int rowChunk = blockIdx.x / 3;
    int col = colChunk * 256 + tid;
#pragma unroll
    for (int c = 0; c < NUM_CLASSES; ++c) lds[c * 256 + tid] = 0.0f;
    __syncthreads();
    int r0 = rowChunk * 256;
    for (int r = 0; r < 256; ++r) {
        int row = r0 + r;
        int lab = labels[row];                       // wave-uniform load
        lds[lab * 256 + tid] += emb[(size_t)row * D_DIM + col];
    }
    __syncthreads();
#pragma unroll
    for (int c = 0; c < NUM_CLASSES; ++c)
        atomicAdd(&ws[WS_CSUM + c * D_DIM + col], lds[c * 256 + tid]);
}

// ---------------- pairwise centroid distances (100 blocks x 64 threads) ----------------
__global__ void k_edist(float* __restrict__ ws) {
    __shared__ float red[64];
    int i = blockIdx.x / NUM_CLASSES;
    int j = blockIdx.x % NUM_CLASSES;
    float ni = fmaxf(ws[WS_CNT + i], 1.0f);
    float nj = fmaxf(ws[WS_CNT + j], 1.0f);
    float s = 0.0f;
    for (int e = threadIdx.x; e < D_DIM; e += 64) {
        float ci = ws[WS_CSUM + i * D_DIM + e] / ni;
        float cj = ws[WS_CSUM + j * D_DIM + e] / nj;
        float d = ci - cj;
        s += d * d;
    }
    red[threadIdx.x] = s;
    __syncthreads();
    for (int off = 32; off; off >>= 1) {
        if (threadIdx.x < off) red[threadIdx.x] += red[threadIdx.x + off];
        __syncthreads();
    }
    if (threadIdx.x == 0) ws[WS_EDIST + blockIdx.x] = sqrtf(red[0] + 1e-12f);
}

// ---------------- TALS: normalize by max, MSE vs topo ----------------
__global__ void k_tals(const float* __restrict__ topo, float* __restrict__ ws) {
    __shared__ float red[128];
    int tid = threadIdx.x;
    float m = 0.0f;
    for (int e = tid; e < NUM_CLASSES * NUM_CLASSES; e += 128)
        m = fmaxf(m, ws[WS_EDIST + e]);
    red[tid] = m;
    __syncthreads();
    for (int off = 64; off; off >>= 1) {
        if (tid < off) red[tid] = fmaxf(red[tid], red[tid + off]);
        __syncthreads();
    }
    float mx = red[0] + 1e-8f;
    __syncthreads();
    float s = 0.0f;
    for (int e = tid; e < NUM_CLASSES * NUM_CLASSES; e += 128) {
        float d = ws[WS_EDIST + e] / mx - topo[e];
        s += d * d;
    }
    red[tid] = s;
    __syncthreads();
    for (int off = 64; off; off >>= 1) {
        if (tid < off) red[tid] += red[tid + off];
        __syncthreads();
    }
    if (tid == 0) ws[WS_TALS] = red[0] / (float)(NUM_CLASSES * NUM_CLASSES);
}

// ---------------- TAML: fused gram-matrix via V_WMMA_F32_16X16X4_F32 ----------------
// One wave32 computes a 32x32 macro-tile of sim = emb @ emb^T as 2x2 WMMA
// sub-tiles with 4 f32 accumulators. Each A/B fragment feeds two WMMAs
// (2x arithmetic intensity vs 1 tile/wave), and interleaving the 4
// accumulators spaces dependent WMMAs 4 instructions apart.
// A-operand (16x4) and B-operand (4x16) layouts are transposes of each
// other, so BOTH operands use the identical per-lane pattern:
//   lane l takes float2 at E[tileBase + (l&15)][k + 2*(l>>4)].
// Upper-triangle macro-tiles only; diagonal macro: sub(0,0)/(1,1) weight 1,
// sub(0,1) weight 2, sub(1,0) skipped. Off-diagonal macro: all weight 2.
__global__ void __launch_bounds__(256)
k_taml(const float* __restrict__ emb,
       const int* __restrict__ labels,
       const float* __restrict__ topo,
       float* __restrict__ ws) {
    const int T2 = B_ROWS / 32;                      // 128 macro-tiles per dim
    int wid = (blockIdx.x * blockDim.x + threadIdx.x) >> 5;
    int bi = wid >> 7;                               // wid / T2
    int bj = wid & (T2 - 1);                         // wid % T2
    if (bi >= T2) return;                            // uniform per wave
    if (bj < bi) return;                             // symmetry: skip lower triangle

    int lane = threadIdx.x & 31;
    int half = lane >> 4;                            // 0 or 1
    int lrow = lane & 15;
    int i0 = bi * 32, j0 = bj * 32;

    const float* pA0 = emb + (size_t)(i0 + lrow) * D_DIM + 2 * half;
    const float* pA1 = pA0 + (size_t)16 * D_DIM;
    const float* pB0 = emb + (size_t)(j0 + lrow) * D_DIM + 2 * half;
    const float* pB1 = pB0 + (size_t)16 * D_DIM;

    v8f acc00 = {0.f,0.f,0.f,0.f,0.f,0.f,0.f,0.f};
    v8f acc01 = {0.f,0.f,0.f,0.f,0.f,0.f,0.f,0.f};
    v8f acc10 = {0.f,0.f,0.f,0.f,0.f,0.f,0.f,0.f};
    v8f acc11 = {0.f,0.f,0.f,0.f,0.f,0.f,0.f,0.f};

#pragma unroll 4
    for (int k = 0; k < D_DIM; k += 4) {
        v2f a0 = *(const v2f*)(pA0 + k);
        v2f a1 = *(const v2f*)(pA1 + k);
        v2f b0 = *(const v2f*)(pB0 + k);
        v2f b1 = *(const v2f*)(pB1 + k);
        acc00 = __builtin_amdgcn_wmma_f32_16x16x4_f32(
            false, a0, false, b0, (short)0, acc00, false, false);
        acc01 = __builtin_amdgcn_wmma_f32_16x16x4_f32(
            false, a0, false, b1, (short)0, acc01, false, false);
        acc10 = __builtin_amdgcn_wmma_f32_16x16x4_f32(
            false, a1, false, b0, (short)0, acc10, false, false);
        acc11 = __builtin_amdgcn_wmma_f32_16x16x4_f32(
            false, a1, false, b1, (short)0, acc11, false, false);
    }

    // Epilogue. C/D layout: VGPR r, lanes 0-15 -> M=r; lanes 16-31 -> M=r+8;
    // N = lane & 15 in both halves. Branchless (la!=lb) masking.
    bool diag = (bi == bj);
    float s = 0.0f;
#pragma unroll
    for (int sub = 0; sub < 4; ++sub) {
        int si = sub >> 1, sj = sub & 1;
        float w;
        if (diag) {
            if (si == 1 && sj == 0) continue;        // mirrored by sub(0,1)
            w = (si == sj) ? 1.0f : 2.0f;
        } else {
            w = 2.0f;
        }
        const v8f& acc = (sub == 0) ? acc00 : (sub == 1) ? acc01
                        : (sub == 2) ? acc10 : acc11;
        int gj = j0 + sj * 16 + lrow;
        int lb = labels[gj];
        float invnj = ws[WS_INVN + gj];
#pragma unroll
        for (int r = 0; r < 8; ++r) {
            int gi = i0 + si * 16 + r + 8 * half;
            int la = labels[gi];
            float sim = acc[r] * invnj * ws[WS_INVN + gi];
            float v = fmaxf(sim - (1.0f - topo[la * NUM_CLASSES + lb]), 0.0f);
            float mask = (la != lb) ? 1.0f : 0.0f;
            s += w * mask * v;
        }
    }
#pragma unroll
    for (int off = 16; off; off >>= 1) s += __shfl_xor(s, off, 32);
    if (lane == 0) atomicAdd(&ws[WS_TAML], s);
}

// ---------------- combine ----------------
__global__ void k_final(const float* __restrict__ ws, float* __restrict__ out) {
    if (threadIdx.x == 0 && blockIdx.x == 0) {
        float ce = ws[WS_CE] / (float)B_ROWS;
        float tals = ws[WS_TALS];
        float sumsq = 0.0f;
#pragma unroll
        for (int c = 0; c < NUM_CLASSES; ++c) {
            float n = ws[WS_CNT + c];
            sumsq += n * n;
        }
        float nvalid = (float)B_ROWS * (float)B_ROWS - sumsq;
        float taml = ws[WS_TAML] / fmaxf(nvalid, 1.0f);
        out[0] = ce + LAM_TALS * tals + LAM_TAML * taml;
        out[1] = ce;
        out[2] = tals;
        out[3] = taml;
    }
}

extern "C" void kernel_launch(void* const* d_in, const int* in_sizes, int n_in,
                              void* d_out, int out_size, void* d_ws, size_t ws_size,
                              hipStream_t stream) {
    const float* logits = (const float*)d_in[0];
    const int*   labels = (const int*)d_in[1];
    const float* emb    = (const float*)d_in[2];
    const float* topo   = (const float*)d_in[3];
    float* ws  = (float*)d_ws;
    float* out = (float*)d_out;

    k_zero<<<(WS_ZERO_N + 255) / 256, 256, 0, stream>>>(ws, WS_ZERO_N);
    k_ce_counts<<<B_ROWS / 256, 256, 0, stream>>>(logits, labels, ws);
    k_invnorm<<<(B_ROWS * 32) / 256, 256, 0, stream>>>(emb, ws);
    k_centsum<<<48, 256, 0, stream>>>(emb, labels, ws);
    k_edist<<<NUM_CLASSES * NUM_CLASSES, 64, 0, stream>>>(ws);
    k_tals<<<1, 128, 0, stream>>>(topo, ws);
    // (128*128 macro-tiles) / (8 waves per 256-thread block)
    k_taml<<<(128 * 128) / 8, 256, 0, stream>>>(emb, labels, topo, ws);
    k_final<<<1, 32, 0, stream>>>(ws, out);
}